// HoughVoting_4707284157035
// MI455X (gfx1250) — compile-verified
//
#include <hip/hip_runtime.h>
#include <math.h>

typedef __attribute__((ext_vector_type(2))) float v2f;
typedef __attribute__((ext_vector_type(8))) float v8f;

#define GXD 272
#define GYD 112
#define GZD 272
#define NVOX (GXD * GYD * GZD)   // 8,286,208
#define NROT 120
#define RESF 0.03f

// pi/120 rounded to float32 (matches np.float32(np.pi / 120))
#define THSTEP 0.02617993877991494f

__device__ __forceinline__ void atomic_add_f32_noret(float* addr, float v) {
    // Relaxed, device-scope fp32 atomic add; result unused so the backend
    // emits the non-returning global_atomic_add_f32 (STOREcnt-tracked).
    (void)__hip_atomic_fetch_add(addr, v, __ATOMIC_RELAXED, __HIP_MEMORY_SCOPE_AGENT);
}

__global__ void __launch_bounds__(256) zero_f32_kernel(float* __restrict__ p, long long n) {
    long long i = (long long)blockIdx.x * blockDim.x + threadIdx.x;
    long long stride = (long long)gridDim.x * blockDim.x;
    for (long long k = i; k < n; k += stride) p[k] = 0.0f;
}

// Block = 16 points. Wave w (of 8) = rotations [16w, 16w+16) (120 padded to 128).
// Per wave: two V_WMMA_F32_16X16X4_F32 produce cx[16x16] and cz[16x16]:
//   cx[m][n] = px[m] - fx[m]*cos(th_n) - fz[m]*sin(th_n)
//   cz[m][n] = pz[m] + fx[m]*sin(th_n) - fz[m]*cos(th_n)
// A layout (16x4 f32): lanes 0-15 hold (K=0,K=1)=(fx,fz) for M=lane;
//                      lanes 16-31 hold (K=2,K=3)=(p,0) for M=lane-16.
// B layout (4x16 f32): lanes 0-15 hold (K=0,K=1) for N=lane;
//                      lanes 16-31 hold (K=2,K=3)=(1,0) for N=lane-16.
// C/D layout: VGPR j -> row M=j (lanes 0-15) / M=j+8 (lanes 16-31), N=lane%16.
__global__ void __launch_bounds__(256) hough_vote_kernel(
    const float* __restrict__ points,  // (N,3)
    const float* __restrict__ xyz,     // (N,3)
    const float* __restrict__ scale,   // (N,3)
    const float* __restrict__ obj,     // (N,)
    float* __restrict__ out)           // [NVOX obj | NVOX rot | NVOX*3 scale]
{
    const int tid  = threadIdx.x;
    const int wave = tid >> 5;        // 0..7 -> rotation tile
    const int lane = tid & 31;
    const int half = lane >> 4;       // 0: lanes 0-15, 1: lanes 16-31
    const int nn   = lane & 15;
    const int pbase = blockIdx.x * 16;
    const int rbase = wave * 16;

    // ---- A operands ----
    const int pa = pbase + nn;
    const float fx = xyz[3 * pa + 0] * scale[3 * pa + 0];
    const float fz = xyz[3 * pa + 2] * scale[3 * pa + 2];
    const float px = points[3 * pa + 0];
    const float pz = points[3 * pa + 2];

    v2f a_cx, a_cz;
    a_cx.x = half ? px : fx;
    a_cx.y = half ? 0.0f : fz;
    a_cz.x = half ? pz : fx;
    a_cz.y = half ? 0.0f : fz;

    // ---- B operands ----
    const float thn = (float)(rbase + nn) * THSTEP;
    const float ct = cosf(thn);
    const float st = sinf(thn);
    v2f b_cx, b_cz;
    b_cx.x = half ? 1.0f : -ct;
    b_cx.y = half ? 0.0f : -st;
    b_cz.x = half ? 1.0f : st;
    b_cz.y = half ? 0.0f : -ct;

    const v8f zero = {};
    // EXEC is all-1s here (no divergence before this point in the wave).
    v8f cxm = __builtin_amdgcn_wmma_f32_16x16x4_f32(
        false, a_cx, false, b_cx, (short)0, zero, false, false);
    v8f czm = __builtin_amdgcn_wmma_f32_16x16x4_f32(
        false, a_cz, false, b_cz, (short)0, zero, false, false);

    float* __restrict__ out_obj   = out;
    float* __restrict__ out_rot   = out + (long long)NVOX;
    float* __restrict__ out_scale = out + 2LL * NVOX;

    const int   r      = rbase + nn;        // this lane's rotation column
    const bool  rvalid = (r < NROT);        // padded rotations 120..127 masked
    const float theta_r = (float)r * THSTEP;
    const float inv_res = 1.0f / RESF;      // constant-folded reciprocal

    if (rvalid) {
#pragma unroll
        for (int j = 0; j < 8; ++j) {
            const int m = j + (half << 3);
            const int p = pbase + m;

            const float cy  = points[3 * p + 1] - xyz[3 * p + 1] * scale[3 * p + 1];
            const float sx  = scale[3 * p + 0];
            const float sy  = scale[3 * p + 1];
            const float sz  = scale[3 * p + 2];
            const float wob = obj[p];

            const float gx = cxm[j] * inv_res;
            const float gy = cy     * inv_res;
            const float gz = czm[j] * inv_res;

            const float x0f = floorf(gx);
            const float y0f = floorf(gy);
            const float z0f = floorf(gz);
            const float wx1 = gx - x0f;
            const float wy1 = gy - y0f;
            const float wz1 = gz - z0f;
            const int ix0 = (int)x0f;
            const int iy0 = (int)y0f;
            const int iz0 = (int)z0f;

#pragma unroll
            for (int dx = 0; dx < 2; ++dx) {
                const int   ix = ix0 + dx;
                const float wx = dx ? wx1 : (1.0f - wx1);
                if (ix < 0 || ix >= GXD) continue;
#pragma unroll
                for (int dy = 0; dy < 2; ++dy) {
                    const int   iy = iy0 + dy;
                    const float wy = dy ? wy1 : (1.0f - wy1);
                    if (iy < 0 || iy >= GYD) continue;
#pragma unroll
                    for (int dz = 0; dz < 2; ++dz) {
                        const int   iz = iz0 + dz;
                        const float wz = dz ? wz1 : (1.0f - wz1);
                        if (iz < 0 || iz >= GZD) continue;

                        const float w = wx * wy * wz * wob;
                        const long long flat = ((long long)ix * GYD + iy) * GZD + iz;
                        atomic_add_f32_noret(out_obj + flat, w);
                        atomic_add_f32_noret(out_rot + flat, w * theta_r);
                        atomic_add_f32_noret(out_scale + 3 * flat + 0, w * sx);
                        atomic_add_f32_noret(out_scale + 3 * flat + 1, w * sy);
                        atomic_add_f32_noret(out_scale + 3 * flat + 2, w * sz);
                    }
                }
            }
        }
    }
}

extern "C" void kernel_launch(void* const* d_in, const int* in_sizes, int n_in,
                              void* d_out, int out_size, void* d_ws, size_t ws_size,
                              hipStream_t stream) {
    (void)n_in; (void)d_ws; (void)ws_size;
    const float* points = (const float*)d_in[0];
    const float* xyz    = (const float*)d_in[1];
    const float* scale  = (const float*)d_in[2];
    const float* obj    = (const float*)d_in[3];
    float* out = (float*)d_out;

    const int npoints = in_sizes[3];          // obj is (N,)

    // 1) Zero the (poisoned) output grids.
    zero_f32_kernel<<<4096, 256, 0, stream>>>(out, (long long)out_size);

    // 2) Vote. One block per 16 points; 8 waves = 8 rotation tiles (128 padded).
    const int nblocks = npoints / 16;
    hough_vote_kernel<<<nblocks, 256, 0, stream>>>(points, xyz, scale, obj, out);
}